// PBDCorrectedVelocityStrategy_43860206026940
// MI455X (gfx1250) — compile-verified
//
#include <hip/hip_runtime.h>
#include <hip/hip_bf16.h>

// ---------------------------------------------------------------------------
// PBD-corrected velocity head for MI455X (gfx1250, wave32)
//   phase 1: x_pred = kp + tau * (hand_tokens @ head_w + head_b)   [WMMA f32]
//   phase 2: 4x Jacobi XPBD edge projection (gather + atomic scatter-add)
//   phase 3: v_eff = (x_final - kp) / tau
// ---------------------------------------------------------------------------

typedef float v2f __attribute__((ext_vector_type(2)));
typedef float v8f __attribute__((ext_vector_type(8)));

#define TAU_MIN_F     0.001f
#define XPBD_MAXC     0.15f
#define XPBD_DENOM    (2.0f + 1e-9f)

// 16 N-rows (cols 3..15 zero-padded) x 256 K, stride padded to 260 words so
// the per-wave ds_load_b64 pattern covers all 64 LDS banks exactly once.
#define BSTRIDE 260

__device__ __forceinline__ void atomic_add_f32(float* p, float v) {
    // lowers to global_atomic_add_f32 on gfx1250
    unsafeAtomicAdd(p, v);
}

// ---------------------------------------------------------------------------
// Kernel 0: zero the delta accumulator
// ---------------------------------------------------------------------------
__global__ void zero_kernel(float* __restrict__ p, int n) {
    int i = blockIdx.x * blockDim.x + threadIdx.x;
    if (i < n) p[i] = 0.0f;
}

// ---------------------------------------------------------------------------
// Kernel 1: head GEMM via V_WMMA_F32_16X16X4_F32 + x_pred
//   one wave per 16-row tile of the (B*N, 256) token matrix
//   A tile: 16x4 f32 (2 VGPR), B tile: 4x16 f32 (2 VGPR, cols 3..15 zero)
//   B operand staged in LDS transposed + zero-padded -> unconditional b64 DS
//   loads, no exec-mask manipulation in the hot loop.
// ---------------------------------------------------------------------------
__global__ __launch_bounds__(256) void head_wmma_kernel(
    const float* __restrict__ ht,      // (B*N, 256)
    const float* __restrict__ W,       // (256, 3)
    const float* __restrict__ bias,    // (3,)
    const float* __restrict__ kp,      // (B*N, 3)
    const float* __restrict__ ts,      // (B,)
    float* __restrict__ x,             // (B*N, 3) out
    int nTiles, int D, int N)
{
    __shared__ float sWT[16 * BSTRIDE];   // [n][k], zero for n >= 3
    __shared__ float sB[3];

    // cooperative fill: n = idx / D, k = idx % D
    for (int idx = threadIdx.x; idx < 16 * 256; idx += blockDim.x) {
        int n = idx >> 8;            // / 256
        int k = idx & 255;           // % 256
        sWT[n * BSTRIDE + k] = (n < 3) ? W[k * 3 + n] : 0.0f;
    }
    if (threadIdx.x < 3) sB[threadIdx.x] = bias[threadIdx.x];
    __syncthreads();

    int wave = (blockIdx.x * blockDim.x + threadIdx.x) >> 5;
    if (wave >= nTiles) return;                 // wave-uniform: EXEC stays all-1s

    int lane = threadIdx.x & 31;
    int half = lane >> 4;                       // 0: lanes 0-15, 1: lanes 16-31
    int l15  = lane & 15;

    long rowA = (long)wave * 16 + l15;          // A-matrix row (M = lane&15)
    const float* arow = ht + rowA * D + 2 * half;
    const float* brow = &sWT[l15 * BSTRIDE + 2 * half];

    v8f acc0 = {};
    v8f acc1 = {};
    #pragma unroll 4
    for (int kb = 0; kb < D; kb += 8) {
        // A: lanes 0-15 hold K=kb,kb+1 ; lanes 16-31 hold K=kb+2,kb+3
        v2f a0 = *(const v2f*)(arow + kb);
        v2f a1 = *(const v2f*)(arow + kb + 4);
        // B: VGPR0 row K=kb+2*half, VGPR1 row K=kb+2*half+1 ; N = lane&15
        v2f b0 = *(const v2f*)(brow + kb);
        v2f b1 = *(const v2f*)(brow + kb + 4);
        acc0 = __builtin_amdgcn_wmma_f32_16x16x4_f32(
                   false, a0, false, b0, (short)0, acc0, false, false);
        acc1 = __builtin_amdgcn_wmma_f32_16x16x4_f32(
                   false, a1, false, b1, (short)0, acc1, false, false);
    }

    // C/D layout: VGPR r holds M = r + 8*half, N = lane&15. Only N<3 is real.
    if (l15 < 3) {
        long rowBase = (long)wave * 16 + half * 8;
        #pragma unroll
        for (int r = 0; r < 8; ++r) {
            long row = rowBase + r;
            int  bidx = (int)(row / N);
            float tau = fmaxf(1.0f - ts[bidx], TAU_MIN_F);
            long idx = row * 3 + l15;
            x[idx] = kp[idx] + tau * (acc0[r] + acc1[r] + sB[l15]);
        }
    }
}

// ---------------------------------------------------------------------------
// Kernel 2: XPBD edge pass (Jacobi): corrections from snapshot x -> delta
// ---------------------------------------------------------------------------
__global__ void edge_kernel(
    const float* __restrict__ x,       // (B*N, 3) snapshot
    const int*   __restrict__ ei,      // (2, E)
    const float* __restrict__ rl,      // (E,)
    float* __restrict__ delta,         // (B*N, 3) accumulator
    int E, int N, int BE)
{
    int t = blockIdx.x * blockDim.x + threadIdx.x;
    if (t >= BE) return;
    int b = t / E;
    int e = t - b * E;
    int s = ei[e];
    int d = ei[E + e];
    int si = (b * N + s) * 3;
    int di = (b * N + d) * 3;

    float dx = x[si + 0] - x[di + 0];
    float dy = x[si + 1] - x[di + 1];
    float dz = x[si + 2] - x[di + 2];
    float dist = sqrtf(dx * dx + dy * dy + dz * dz);
    float C    = dist - rl[e];
    float scal = (-C / XPBD_DENOM) / (dist + 1e-9f);

    float cx = fminf(fmaxf(scal * dx, -XPBD_MAXC), XPBD_MAXC);
    float cy = fminf(fmaxf(scal * dy, -XPBD_MAXC), XPBD_MAXC);
    float cz = fminf(fmaxf(scal * dz, -XPBD_MAXC), XPBD_MAXC);

    atomic_add_f32(&delta[si + 0],  cx);
    atomic_add_f32(&delta[si + 1],  cy);
    atomic_add_f32(&delta[si + 2],  cz);
    atomic_add_f32(&delta[di + 0], -cx);
    atomic_add_f32(&delta[di + 1], -cy);
    atomic_add_f32(&delta[di + 2], -cz);
}

// ---------------------------------------------------------------------------
// Kernel 3: apply delta to x and re-zero delta for the next iteration
// ---------------------------------------------------------------------------
__global__ void apply_kernel(float* __restrict__ x, float* __restrict__ delta, int n) {
    int i = blockIdx.x * blockDim.x + threadIdx.x;
    if (i < n) {
        x[i] += delta[i];
        delta[i] = 0.0f;
    }
}

// ---------------------------------------------------------------------------
// Kernel 4: fold in last delta and emit v_eff = (x_final - kp) / tau
// ---------------------------------------------------------------------------
__global__ void final_kernel(
    const float* __restrict__ x, const float* __restrict__ delta,
    const float* __restrict__ kp, const float* __restrict__ ts,
    float* __restrict__ out, int n, int perBatch)
{
    int i = blockIdx.x * blockDim.x + threadIdx.x;
    if (i < n) {
        int b = i / perBatch;
        float tau = fmaxf(1.0f - ts[b], TAU_MIN_F);
        out[i] = (x[i] + delta[i] - kp[i]) / tau;
    }
}

// ---------------------------------------------------------------------------
// launcher
// ---------------------------------------------------------------------------
extern "C" void kernel_launch(void* const* d_in, const int* in_sizes, int n_in,
                              void* d_out, int out_size, void* d_ws, size_t ws_size,
                              hipStream_t stream) {
    const float* kp   = (const float*)d_in[0];   // (B,N,3)
    const float* ts   = (const float*)d_in[1];   // (B,)
    const float* ht   = (const float*)d_in[2];   // (B,N,D)
    const float* W    = (const float*)d_in[3];   // (D,3)
    const float* bias = (const float*)d_in[4];   // (3,)
    const int*   ei   = (const int*)d_in[5];     // (2,E)
    const float* rl   = (const float*)d_in[6];   // (E,)

    const int B   = in_sizes[1];
    const int E   = in_sizes[6];
    const int BN3 = in_sizes[0];                 // B*N*3
    const int BN  = BN3 / 3;
    const int N   = BN / B;
    const int D   = in_sizes[2] / BN;

    float* xbuf  = (float*)d_ws;                 // B*N*3 floats
    float* delta = xbuf + BN3;                   // B*N*3 floats
    (void)ws_size;

    const int T = 256;
    const int nTiles = (BN + 15) / 16;           // 16 rows per wave (BN % 16 == 0 here)
    const int gVec   = (BN3 + T - 1) / T;
    const int BE     = B * E;
    const int gEdge  = (BE + T - 1) / T;
    const int gWmma  = (nTiles * 32 + T - 1) / T;

    zero_kernel<<<gVec, T, 0, stream>>>(delta, BN3);
    head_wmma_kernel<<<gWmma, T, 0, stream>>>(ht, W, bias, kp, ts, xbuf, nTiles, D, N);

    const int XPBD_ITERS = 4;
    for (int it = 0; it < XPBD_ITERS; ++it) {
        edge_kernel<<<gEdge, T, 0, stream>>>(xbuf, ei, rl, delta, E, N, BE);
        if (it < XPBD_ITERS - 1) {
            apply_kernel<<<gVec, T, 0, stream>>>(xbuf, delta, BN3);
        } else {
            final_kernel<<<gVec, T, 0, stream>>>(xbuf, delta, kp, ts,
                                                 (float*)d_out, BN3, N * 3);
        }
    }
}